// HubDetectionDiscriminator_50508815401437
// MI455X (gfx1250) — compile-verified
//
#include <hip/hip_runtime.h>
#include <hip/hip_bf16.h>
#include <math.h>

typedef __attribute__((ext_vector_type(2))) float v2f;
typedef __attribute__((ext_vector_type(8))) float v8f;
typedef __attribute__((ext_vector_type(4))) unsigned int u32x4;
typedef __attribute__((ext_vector_type(8))) unsigned int u32x8;

#define NGRAPH 64
#define CDIV(a, b) (((a) + (b) - 1) / (b))

// ---------------------------------------------------------------------------
// utility kernels
// ---------------------------------------------------------------------------
__global__ void k_fill(float* __restrict__ p, float v, int n) {
  int i = blockIdx.x * blockDim.x + threadIdx.x;
  if (i < n) p[i] = v;
}

__global__ void k_init_bias(float* __restrict__ out, const float* __restrict__ bias, int total, int P) {
  int i = blockIdx.x * blockDim.x + threadIdx.x;
  if (i < total) out[i] = bias[i % P];
}

__device__ inline void atomicMaxF(float* addr, float val) {
  unsigned* ua = reinterpret_cast<unsigned*>(addr);
  unsigned cur = *(volatile unsigned*)ua;
  while (__uint_as_float(cur) < val) {
    unsigned prev = atomicCAS(ua, cur, __float_as_uint(val));
    if (prev == cur) break;
    cur = prev;
  }
}

__device__ inline float lrelu02(float x) { return x > 0.f ? x : 0.2f * x; }

// ---------------------------------------------------------------------------
// Tensor Data Mover: DMA one 16 x ROWLEN f32 tile (contiguous, row-major)
// from global memory into LDS.  D# per CDNA5 ISA ch.8:
//  group0: count=1 | lds_addr | global_addr[56:0] | type=2
//  group1: data_size=4B, tensor_dim0=ROWLEN, tensor_dim1=16,
//          tile_dim0=ROWLEN, tile_dim1=16, stride0=ROWLEN
// ---------------------------------------------------------------------------
__device__ inline void tdm_load_tile16(const float* gsrc, void* lds_dst, int row_len) {
  unsigned long long ga = (unsigned long long)(uintptr_t)gsrc;
  unsigned int lds_off = (unsigned int)(uintptr_t)lds_dst;  // low 32b of generic = LDS offset
  u32x4 g0;
  g0[0] = 1u;                                        // count = 1 valid descriptor
  g0[1] = lds_off;                                   // lds_addr [63:32]
  g0[2] = (unsigned int)ga;                          // global_addr lo
  g0[3] = (unsigned int)(ga >> 32) | (2u << 30);     // global_addr hi | type=2 (image)
  unsigned rl = (unsigned)row_len;
  u32x8 g1;
  g1[0] = 2u << 16;                 // workgroup_mask=0, data_size=2 (4 bytes)
  g1[1] = (rl & 0xFFFFu) << 16;     // tensor_dim0[15:0]
  g1[2] = (rl >> 16) | (16u << 16); // tensor_dim0[31:16] | tensor_dim1[15:0]=16
  g1[3] = (rl & 0xFFFFu) << 16;     // tensor_dim1[31:16]=0 | tile_dim0=row_len
  g1[4] = 16u;                      // tile_dim1=16, tile_dim2=0
  g1[5] = rl;                       // tensor_dim0_stride[31:0]
  g1[6] = 0u;                       // stride0 hi | tensor_dim1_stride lo
  g1[7] = 0u;
  asm volatile("tensor_load_to_lds %0, %1" :: "s"(g0), "s"(g1) : "memory");
}

// ---------------------------------------------------------------------------
// Persistent WMMA f32 GEMM: out[rows,P] = A[rows,K] @ W[K,P] (+bias)(+relu)
//  - rows must be a multiple of 16 (N = 50000 = 3125*16)
//  - blockDim.x = 32*(P/16); wave w owns the 16-col stripe [16w,16w+16)
//  - B stripe lives in VGPRs (K/4 v2f frags), loaded once per block
//  - A tiles DMA'd global->LDS by the TDM, double buffered on TENSORcnt
// Fragment layouts per ISA 7.12.2:
//  A 16x4: lanes 0-15 K={k0,k0+1}, lanes 16-31 K={k0+2,k0+3}, M=lane&15
//  B 4x16: same K split, N=lane&15;  C/D 16x16: VGPR i -> M=i+8*(lane>>4)
// ---------------------------------------------------------------------------
template <int K, int P, bool BIAS, bool RELU>
__global__ void wmma_gemm(const float* __restrict__ A, const float* __restrict__ W,
                          const float* __restrict__ bias, float* __restrict__ out, int rows) {
  __shared__ __align__(16) float sA[2][16 * K];
  const int ntiles = rows >> 4;
  const int lane = threadIdx.x & 31;
  const int wave = threadIdx.x >> 5;
  const int m    = lane & 15;
  const int kk   = (lane >> 4) * 2;   // 0 or 2
  const int col  = wave * 16 + m;
  const int half = lane >> 4;

  // B stripe resident in registers for the whole kernel
  v2f b[K / 4];
#pragma unroll
  for (int j = 0; j < K / 4; ++j) {
    b[j].x = W[(size_t)(4 * j + kk) * P + col];
    b[j].y = W[(size_t)(4 * j + kk + 1) * P + col];
  }
  const float bb = BIAS ? bias[col] : 0.f;

  int t = blockIdx.x;
  if (t >= ntiles) return;
  int cur = 0;
  if (threadIdx.x == 0) tdm_load_tile16(A + (size_t)t * 16 * K, &sA[0][0], K);

  for (; t < ntiles; t += gridDim.x) {
    if (threadIdx.x == 0) __builtin_amdgcn_s_wait_tensorcnt(0);
    __syncthreads();                              // sA[cur] ready for all waves
    int tn = t + (int)gridDim.x;
    if (tn < ntiles && threadIdx.x == 0)
      tdm_load_tile16(A + (size_t)tn * 16 * K, &sA[cur ^ 1][0], K);  // prefetch

    const float* As = &sA[cur][0];
    v8f c = {};
#pragma unroll
    for (int j = 0; j < K / 4; ++j) {
      v2f a = *(const v2f*)(As + m * K + 4 * j + kk);   // ds_load_b64
      c = __builtin_amdgcn_wmma_f32_16x16x4_f32(false, a, false, b[j], (short)0, c, false, false);
    }
#pragma unroll
    for (int i = 0; i < 8; ++i) {
      float v = c[i] + bb;
      if (RELU) v = fmaxf(v, 0.f);
      out[(size_t)(t * 16 + i + 8 * half) * P + col] = v;
    }
    cur ^= 1;
    __syncthreads();                              // done reading before next DMA overwrites
  }
}

// ---------------------------------------------------------------------------
// degree / GCN
// ---------------------------------------------------------------------------
__global__ void k_deg(const int* __restrict__ ei, float* __restrict__ deg, int E, int E2) {
  int t = blockIdx.x * blockDim.x + threadIdx.x;
  if (t >= E2) return;
  int dst = (t < E) ? ei[E + t] : (t - E);
  atomicAdd(&deg[dst], 1.0f);
}

__global__ void k_dinv(float* __restrict__ deg, int n) {
  int i = blockIdx.x * blockDim.x + threadIdx.x;
  if (i >= n) return;
  float d = deg[i];
  deg[i] = d > 0.f ? rsqrtf(fmaxf(d, 1e-12f)) : 0.f;
}

// one wave per edge; lane covers 4 features
__global__ void k_gcn_agg(const int* __restrict__ ei, const float* __restrict__ xw,
                          const float* __restrict__ dinv, float* __restrict__ agg, int E, int E2) {
  int w = (blockIdx.x * blockDim.x + threadIdx.x) >> 5;
  int lane = threadIdx.x & 31;
  if (w >= E2) return;
  int src = (w < E) ? ei[w]     : (w - E);
  int dst = (w < E) ? ei[E + w] : (w - E);
  float nrm = dinv[src] * dinv[dst];
  const float4 v = *(const float4*)(xw + (size_t)src * 128 + lane * 4);
  float* d = agg + (size_t)dst * 128 + lane * 4;
  atomicAdd(d + 0, v.x * nrm); atomicAdd(d + 1, v.y * nrm);
  atomicAdd(d + 2, v.z * nrm); atomicAdd(d + 3, v.w * nrm);
}

// ---------------------------------------------------------------------------
// GraphNorm (batch-segment mean/var via L2 atomics)
// ---------------------------------------------------------------------------
__global__ void k_cnt(const int* __restrict__ batch, float* __restrict__ cnt, int n) {
  int i = blockIdx.x * blockDim.x + threadIdx.x;
  if (i < n) atomicAdd(&cnt[batch[i]], 1.0f);
}

__global__ void k_gn_mean(const float* __restrict__ h, const int* __restrict__ batch,
                          float* __restrict__ mean, int n) {
  int w = (blockIdx.x * blockDim.x + threadIdx.x) >> 5;
  int lane = threadIdx.x & 31;
  if (w >= n) return;
  int g = batch[w];
  const float4 v = *(const float4*)(h + (size_t)w * 128 + lane * 4);
  float* d = mean + g * 128 + lane * 4;
  atomicAdd(d + 0, v.x); atomicAdd(d + 1, v.y); atomicAdd(d + 2, v.z); atomicAdd(d + 3, v.w);
}

__global__ void k_div_cnt(float* __restrict__ p, const float* __restrict__ cnt, int total, int P) {
  int i = blockIdx.x * blockDim.x + threadIdx.x;
  if (i < total) p[i] /= fmaxf(cnt[i / P], 1.0f);
}

__global__ void k_gn_var(const float* __restrict__ h, const int* __restrict__ batch,
                         const float* __restrict__ mean, const float* __restrict__ ms,
                         float* __restrict__ var, int n) {
  int w = (blockIdx.x * blockDim.x + threadIdx.x) >> 5;
  int lane = threadIdx.x & 31;
  if (w >= n) return;
  int g = batch[w];
  int f0 = lane * 4;
  const float4 v = *(const float4*)(h + (size_t)w * 128 + f0);
  float s0 = v.x - ms[f0 + 0] * mean[g * 128 + f0 + 0];
  float s1 = v.y - ms[f0 + 1] * mean[g * 128 + f0 + 1];
  float s2 = v.z - ms[f0 + 2] * mean[g * 128 + f0 + 2];
  float s3 = v.w - ms[f0 + 3] * mean[g * 128 + f0 + 3];
  float* d = var + g * 128 + f0;
  atomicAdd(d + 0, s0 * s0); atomicAdd(d + 1, s1 * s1);
  atomicAdd(d + 2, s2 * s2); atomicAdd(d + 3, s3 * s3);
}

// x_res += relu(w*(h - ms*mean)*rsqrt(var+eps) + b)
__global__ void k_gn_apply(const float* __restrict__ h, const int* __restrict__ batch,
                           const float* __restrict__ mean, const float* __restrict__ var,
                           const float* __restrict__ w, const float* __restrict__ b,
                           const float* __restrict__ ms, float* __restrict__ xres, int n) {
  int i = blockIdx.x * blockDim.x + threadIdx.x;
  if (i >= n * 128) return;
  int node = i >> 7, f = i & 127;
  int g = batch[node];
  float sub = h[i] - ms[f] * mean[g * 128 + f];
  float y = w[f] * sub * rsqrtf(var[g * 128 + f] + 1e-5f) + b[f];
  xres[i] += fmaxf(y, 0.f);
}

// ---------------------------------------------------------------------------
// GAT
// ---------------------------------------------------------------------------
__global__ void k_att_scores(const float* __restrict__ hm, const float* __restrict__ asrc,
                             const float* __restrict__ adst, float* __restrict__ es,
                             float* __restrict__ ed, int n) {
  int i = blockIdx.x * blockDim.x + threadIdx.x;
  if (i >= n * 4) return;
  int node = i >> 2, h = i & 3;
  const float* p = hm + (size_t)node * 128 + h * 32;
  float s1 = 0.f, s2 = 0.f;
#pragma unroll 8
  for (int c = 0; c < 32; ++c) { float v = p[c]; s1 += v * asrc[h * 32 + c]; s2 += v * adst[h * 32 + c]; }
  es[i] = s1; ed[i] = s2;
}

__global__ void k_att_max(const int* __restrict__ ei, const float* __restrict__ es,
                          const float* __restrict__ ed, float* __restrict__ m, int E, int E2) {
  int i = blockIdx.x * blockDim.x + threadIdx.x;
  if (i >= E2 * 4) return;
  int t = i >> 2, h = i & 3;
  int src = (t < E) ? ei[t]     : (t - E);
  int dst = (t < E) ? ei[E + t] : (t - E);
  float e = lrelu02(es[src * 4 + h] + ed[dst * 4 + h]);
  atomicMaxF(&m[dst * 4 + h], e);
}

__global__ void k_att_sum(const int* __restrict__ ei, const float* __restrict__ es,
                          const float* __restrict__ ed, const float* __restrict__ m,
                          float* __restrict__ s, int E, int E2) {
  int i = blockIdx.x * blockDim.x + threadIdx.x;
  if (i >= E2 * 4) return;
  int t = i >> 2, h = i & 3;
  int src = (t < E) ? ei[t]     : (t - E);
  int dst = (t < E) ? ei[E + t] : (t - E);
  float e = lrelu02(es[src * 4 + h] + ed[dst * 4 + h]);
  atomicAdd(&s[dst * 4 + h], __expf(e - m[dst * 4 + h]));
}

// one wave per edge; lane covers 4 features (same head within a lane)
__global__ void k_gat_agg(const int* __restrict__ ei, const float* __restrict__ hm,
                          const float* __restrict__ es, const float* __restrict__ ed,
                          const float* __restrict__ m, const float* __restrict__ s,
                          float* __restrict__ agg, int E, int E2) {
  int w = (blockIdx.x * blockDim.x + threadIdx.x) >> 5;
  int lane = threadIdx.x & 31;
  if (w >= E2) return;
  int src = (w < E) ? ei[w]     : (w - E);
  int dst = (w < E) ? ei[E + w] : (w - E);
  int f0 = lane * 4, h = lane >> 3;
  float e = lrelu02(es[src * 4 + h] + ed[dst * 4 + h]);
  float alpha = __expf(e - m[dst * 4 + h]) / (s[dst * 4 + h] + 1e-16f);
  const float4 v = *(const float4*)(hm + (size_t)src * 128 + f0);
  float* d = agg + (size_t)dst * 128 + f0;
  atomicAdd(d + 0, v.x * alpha); atomicAdd(d + 1, v.y * alpha);
  atomicAdd(d + 2, v.z * alpha); atomicAdd(d + 3, v.w * alpha);
}

// ---------------------------------------------------------------------------
// node scores, pooling, graph MLP
// ---------------------------------------------------------------------------
__global__ void k_node_scores(const float* __restrict__ n1, const float* __restrict__ nw2,
                              const float* __restrict__ nb2, float* __restrict__ out, int n) {
  int i = blockIdx.x * blockDim.x + threadIdx.x;
  if (i >= n) return;
  float acc = nb2[0];
  const float* p = n1 + (size_t)i * 64;
#pragma unroll 8
  for (int j = 0; j < 64; ++j) acc += p[j] * nw2[j];
  out[i] = 1.f / (1.f + __expf(-acc));
}

__global__ void k_pool_init(float* __restrict__ gf) {
  int i = blockIdx.x * blockDim.x + threadIdx.x;
  if (i < NGRAPH * 256) gf[i] = ((i & 255) < 128) ? 0.f : -3.0e38f;
}

__global__ void k_pool(const float* __restrict__ x, const int* __restrict__ batch,
                       float* __restrict__ gf, int n) {
  int w = (blockIdx.x * blockDim.x + threadIdx.x) >> 5;
  int lane = threadIdx.x & 31;
  if (w >= n) return;
  int g = batch[w];
  int f0 = lane * 4;
  const float4 v = *(const float4*)(x + (size_t)w * 128 + f0);
  float* gm = gf + g * 256 + f0;
  float* gx = gf + g * 256 + 128 + f0;
  atomicAdd(gm + 0, v.x); atomicAdd(gm + 1, v.y); atomicAdd(gm + 2, v.z); atomicAdd(gm + 3, v.w);
  atomicMaxF(gx + 0, v.x); atomicMaxF(gx + 1, v.y); atomicMaxF(gx + 2, v.z); atomicMaxF(gx + 3, v.w);
}

__global__ void k_pool_fin(float* __restrict__ gf, const float* __restrict__ cnt) {
  int i = blockIdx.x * blockDim.x + threadIdx.x;
  if (i >= NGRAPH * 128) return;
  int g = i >> 7, f = i & 127;
  gf[g * 256 + f] /= fmaxf(cnt[g], 1.0f);
}

__global__ void k_dense(const float* __restrict__ in, const float* __restrict__ W,
                        const float* __restrict__ b, float* __restrict__ out,
                        int rows, int K, int P, int do_relu) {
  int i = blockIdx.x * blockDim.x + threadIdx.x;
  if (i >= rows * P) return;
  int r = i / P, j = i % P;
  float acc = b[j];
  for (int k = 0; k < K; ++k) acc += in[(size_t)r * K + k] * W[(size_t)k * P + j];
  out[i] = do_relu ? fmaxf(acc, 0.f) : acc;
}

// ---------------------------------------------------------------------------
// launcher
// ---------------------------------------------------------------------------
extern "C" void kernel_launch(void* const* d_in, const int* in_sizes, int n_in,
                              void* d_out, int out_size, void* d_ws, size_t ws_size,
                              hipStream_t stream) {
  const float* x_in = (const float*)d_in[0];
  const int*   ei   = (const int*)d_in[1];
  const int*   batch= (const int*)d_in[2];
  const float* Wp = (const float*)d_in[3];  const float* bp = (const float*)d_in[4];
  const float* Wg = (const float*)d_in[5];  const float* bg = (const float*)d_in[6];
  const float* Wa1=(const float*)d_in[7];   const float* as1=(const float*)d_in[8];
  const float* ad1=(const float*)d_in[9];   const float* ba1=(const float*)d_in[10];
  const float* Wa2=(const float*)d_in[11];  const float* as2=(const float*)d_in[12];
  const float* ad2=(const float*)d_in[13];  const float* ba2=(const float*)d_in[14];
  const float* gnw[3] = {(const float*)d_in[15], (const float*)d_in[18], (const float*)d_in[21]};
  const float* gnb[3] = {(const float*)d_in[16], (const float*)d_in[19], (const float*)d_in[22]};
  const float* gns[3] = {(const float*)d_in[17], (const float*)d_in[20], (const float*)d_in[23]};
  const float* hw1=(const float*)d_in[24];  const float* hb1=(const float*)d_in[25];
  const float* hw2=(const float*)d_in[26];  const float* hb2=(const float*)d_in[27];
  const float* hw3=(const float*)d_in[28];  const float* hb3=(const float*)d_in[29];
  const float* nw1=(const float*)d_in[30];  const float* nb1=(const float*)d_in[31];
  const float* nw2=(const float*)d_in[32];  const float* nb2=(const float*)d_in[33];

  const int n  = in_sizes[0] / 128;
  const int E  = in_sizes[1] / 2;
  const int E2 = E + n;
  const int ntiles = n / 16;
  const int gemm_blocks = ntiles < 1024 ? ntiles : 1024;

  // output layout: logits[G*2] | scores[n] | graph_features[G*256] | x[n*128]
  float* out    = (float*)d_out;
  float* logits = out;
  float* scores = out + NGRAPH * 2;
  float* gfeat  = scores + n;
  float* xcur   = gfeat + NGRAPH * 256;   // lives in d_out; residual accumulator

  // workspace layout
  float* ws      = (float*)d_ws;
  float* buf_xw  = ws;                               // n*128
  float* buf_agg = buf_xw + (size_t)n * 128;         // n*128
  float* deg     = buf_agg + (size_t)n * 128;        // n (becomes dinv)
  float* cnt     = deg + n;                          // G
  float* es      = cnt + NGRAPH;                     // n*4
  float* ed      = es + (size_t)n * 4;               // n*4
  float* mM      = ed + (size_t)n * 4;               // n*4
  float* sS      = mM + (size_t)n * 4;               // n*4
  float* mean    = sS + (size_t)n * 4;               // G*128
  float* var     = mean + NGRAPH * 128;              // G*128
  float* z1      = var + NGRAPH * 128;               // G*128
  float* z2      = z1 + NGRAPH * 128;                // G*64

  auto graphnorm = [&](const float* h, int layer) {
    k_fill<<<CDIV(NGRAPH * 128, 256), 256, 0, stream>>>(mean, 0.f, NGRAPH * 128);
    k_fill<<<CDIV(NGRAPH * 128, 256), 256, 0, stream>>>(var, 0.f, NGRAPH * 128);
    k_gn_mean<<<CDIV(n * 32, 256), 256, 0, stream>>>(h, batch, mean, n);
    k_div_cnt<<<CDIV(NGRAPH * 128, 256), 256, 0, stream>>>(mean, cnt, NGRAPH * 128, 128);
    k_gn_var<<<CDIV(n * 32, 256), 256, 0, stream>>>(h, batch, mean, gns[layer], var, n);
    k_div_cnt<<<CDIV(NGRAPH * 128, 256), 256, 0, stream>>>(var, cnt, NGRAPH * 128, 128);
    k_gn_apply<<<CDIV(n * 128, 256), 256, 0, stream>>>(h, batch, mean, var, gnw[layer],
                                                       gnb[layer], gns[layer], xcur, n);
  };

  auto gat_layer = [&](const float* Wa, const float* as, const float* ad, const float* ba, int layer) {
    wmma_gemm<128, 128, false, false><<<gemm_blocks, 256, 0, stream>>>(xcur, Wa, nullptr, buf_xw, n);
    k_att_scores<<<CDIV(n * 4, 256), 256, 0, stream>>>(buf_xw, as, ad, es, ed, n);
    k_fill<<<CDIV(n * 4, 256), 256, 0, stream>>>(mM, -3.0e38f, n * 4);
    k_att_max<<<CDIV(E2 * 4, 256), 256, 0, stream>>>(ei, es, ed, mM, E, E2);
    k_fill<<<CDIV(n * 4, 256), 256, 0, stream>>>(sS, 0.f, n * 4);
    k_att_sum<<<CDIV(E2 * 4, 256), 256, 0, stream>>>(ei, es, ed, mM, sS, E, E2);
    k_init_bias<<<CDIV(n * 128, 256), 256, 0, stream>>>(buf_agg, ba, n * 128, 128);
    k_gat_agg<<<CDIV(E2 * 32, 256), 256, 0, stream>>>(ei, buf_xw, es, ed, mM, sS, buf_agg, E, E2);
    graphnorm(buf_agg, layer);
  };

  // x = x @ Wp + bp  (written straight into d_out's x slice)
  wmma_gemm<128, 128, true, false><<<gemm_blocks, 256, 0, stream>>>(x_in, Wp, bp, xcur, n);

  // per-graph node counts (constant across layers)
  k_fill<<<1, NGRAPH, 0, stream>>>(cnt, 0.f, NGRAPH);
  k_cnt<<<CDIV(n, 256), 256, 0, stream>>>(batch, cnt, n);

  // degree -> D^-1/2 (self-loops included)
  k_fill<<<CDIV(n, 256), 256, 0, stream>>>(deg, 0.f, n);
  k_deg<<<CDIV(E2, 256), 256, 0, stream>>>(ei, deg, E, E2);
  k_dinv<<<CDIV(n, 256), 256, 0, stream>>>(deg, n);

  // --- GCN layer ---
  wmma_gemm<128, 128, false, false><<<gemm_blocks, 256, 0, stream>>>(xcur, Wg, nullptr, buf_xw, n);
  k_init_bias<<<CDIV(n * 128, 256), 256, 0, stream>>>(buf_agg, bg, n * 128, 128);
  k_gcn_agg<<<CDIV(E2 * 32, 256), 256, 0, stream>>>(ei, buf_xw, deg, buf_agg, E, E2);
  graphnorm(buf_agg, 0);

  // --- GAT layers ---
  gat_layer(Wa1, as1, ad1, ba1, 1);
  gat_layer(Wa2, as2, ad2, ba2, 2);

  // --- node hub scores ---
  wmma_gemm<128, 64, true, true><<<gemm_blocks, 128, 0, stream>>>(xcur, nw1, nb1, buf_xw, n);
  k_node_scores<<<CDIV(n, 256), 256, 0, stream>>>(buf_xw, nw2, nb2, scores, n);

  // --- graph pooling (mean | max) ---
  k_pool_init<<<CDIV(NGRAPH * 256, 256), 256, 0, stream>>>(gfeat);
  k_pool<<<CDIV(n * 32, 256), 256, 0, stream>>>(xcur, batch, gfeat, n);
  k_pool_fin<<<CDIV(NGRAPH * 128, 256), 256, 0, stream>>>(gfeat, cnt);

  // --- graph MLP (tiny; scalar kernels) ---
  k_dense<<<CDIV(NGRAPH * 128, 128), 128, 0, stream>>>(gfeat, hw1, hb1, z1, NGRAPH, 256, 128, 1);
  k_dense<<<CDIV(NGRAPH * 64, 128), 128, 0, stream>>>(z1, hw2, hb2, z2, NGRAPH, 128, 64, 1);
  k_dense<<<1, NGRAPH * 2, 0, stream>>>(z2, hw3, hb3, logits, NGRAPH, 64, 2, 0);
}